// CustomMaskedConv1D_82970178224746
// MI455X (gfx1250) — compile-verified
//
#include <hip/hip_runtime.h>

#define BATCH 4
#define LSEQ 4096
#define CIN 1024
#define G 16
#define IPG 64
#define OPG 64
#define KTAP 5
#define ROWS_PB 128              // output rows per block = 8 waves * 16
#define AROWS (ROWS_PB + 4)      // staged input rows incl. halo (132)
#define ASTRIDE 68               // padded LDS row stride (floats) -> conflict-free b64 A loads
#define BPAIR_STRIDE 160         // floats per kk-pair block (128 data + 32 pad -> disjoint bank halves)
#define NPAIR (KTAP * IPG / 2)   // 160 kk-pairs
#define AS_FLOATS (AROWS * ASTRIDE)       // 8976
#define BS_FLOATS (NPAIR * BPAIR_STRIDE)  // 25600
#define LDS_BYTES ((AS_FLOATS + BS_FLOATS) * 4)  // 138304

typedef __attribute__((ext_vector_type(2))) float v2f;
typedef __attribute__((ext_vector_type(8))) float v8f;

__global__ __launch_bounds__(256) void masked_gconv_wmma(
    const float* __restrict__ inp, const int* __restrict__ pos,
    const float* __restrict__ ker, float* __restrict__ out) {
  extern __shared__ float smem[];
  float* As = smem;               // input tile  [AROWS][ASTRIDE]
  float* Bs = smem + AS_FLOATS;   // kernel tile, pair-interleaved

  const int tid  = threadIdx.x;
  const int lane = tid & 31;
  const int wv   = tid >> 5;

  const int mb = blockIdx.x & 31;
  const int g  = (blockIdx.x >> 5) & 15;
  const int b  = blockIdx.x >> 9;
  const int l0 = mb * ROWS_PB;

  // ---- stage input tile (AROWS x IPG floats) into LDS, coalesced float2 ----
  {
    const float* src = inp + ((size_t)b * LSEQ) * CIN + g * IPG;
    for (int e = tid; e < AROWS * (IPG / 2); e += 256) {
      int row = e >> 5;                 // 0..131  (32 float2 per row)
      int ch2 = (e & 31) << 1;          // 0..62
      int gr  = l0 - 2 + row;
      gr = gr < 0 ? 0 : (gr > LSEQ - 1 ? LSEQ - 1 : gr);  // clamp; mask zeroes halo
      const float2 v = *(const float2*)(src + (size_t)gr * CIN + ch2);
      As[row * ASTRIDE + ch2]     = v.x;
      As[row * ASTRIDE + ch2 + 1] = v.y;
    }
  }

  // ---- stage kernel[g] into LDS as B[kk][o], kk = k*IPG + i,
  //      pair-interleaved: Bs[(kk>>1)*BPAIR_STRIDE + o*2 + (kk&1)] ----
  {
    const float* kg = ker + (size_t)g * OPG * IPG * KTAP;   // 20480 floats
    for (int e4 = tid; e4 < (OPG * IPG * KTAP / 4); e4 += 256) {
      const float4 v = *(const float4*)(kg + e4 * 4);
      float vv[4] = {v.x, v.y, v.z, v.w};
#pragma unroll
      for (int j = 0; j < 4; ++j) {
        int e = e4 * 4 + j;             // e = o*320 + i*5 + k
        int k = e % KTAP;
        int i = (e / KTAP) & (IPG - 1);
        int o = e / (IPG * KTAP);
        int kk = k * IPG + i;
        Bs[(kk >> 1) * BPAIR_STRIDE + (o << 1) + (kk & 1)] = vv[j];
      }
    }
  }

  __syncthreads();

  // ---- per-lane position masks: row = lane&15 matches WMMA A layout ----
  const int rloc = (wv << 4) + (lane & 15);   // local output row of this lane
  const int r    = l0 + rloc;
  const int pc   = pos[b * LSEQ + r];
  float mk[KTAP];
#pragma unroll
  for (int k = 0; k < KTAP; ++k) {
    int rp = r + k - 2;
    int pk = (rp >= 0 && rp < LSEQ) ? pos[b * LSEQ + rp] : (int)0x80000000;
    mk[k] = (pk == pc + (k - 2)) ? 1.0f : 0.0f;
  }

  // ---- WMMA main loop: 5 taps x 16 chunks x 4 N-tiles = 320 wmma/wave ----
  v8f acc[4] = {v8f{}, v8f{}, v8f{}, v8f{}};
  const int hi    = lane >> 4;
  const int abase = rloc * ASTRIDE + (hi << 1);                 // +k*ASTRIDE +i0
  const int bbase = hi * BPAIR_STRIDE + ((lane & 15) << 1);     // +pair*stride +nt*32

  for (int k = 0; k < KTAP; ++k) {
    const float m = mk[k];
    const float* Ak = As + abase + k * ASTRIDE;   // LDS row = rloc + k (halo offset folded)
#pragma unroll 4
    for (int i0 = 0; i0 < IPG; i0 += 4) {
      v2f a;
      {
        const float2 t = *(const float2*)(Ak + i0);
        a.x = t.x * m;
        a.y = t.y * m;
      }
      const int kk0 = k * IPG + i0;
      const float* Bk = Bs + (kk0 >> 1) * BPAIR_STRIDE + bbase;
#pragma unroll
      for (int nt = 0; nt < 4; ++nt) {
        v2f bf = *(const v2f*)(Bk + nt * 32);
        acc[nt] = __builtin_amdgcn_wmma_f32_16x16x4_f32(
            /*neg_a=*/false, a, /*neg_b=*/false, bf,
            /*c_mod=*/(short)0, acc[nt], /*reuse_a=*/false, /*reuse_b=*/false);
      }
    }
  }

  // ---- store: C/D layout -> out[b][row][g][o], 64B-contiguous per half-wave ----
  const int obase = lane & 15;
#pragma unroll
  for (int nt = 0; nt < 4; ++nt) {
#pragma unroll
    for (int v = 0; v < 8; ++v) {
      int row = l0 + (wv << 4) + v + (hi << 3);
      int o   = nt * 16 + obase;
      out[(((size_t)b * LSEQ + row) * G + g) * OPG + o] = acc[nt][v];
    }
  }
}

extern "C" void kernel_launch(void* const* d_in, const int* in_sizes, int n_in,
                              void* d_out, int out_size, void* d_ws, size_t ws_size,
                              hipStream_t stream) {
  const float* inp = (const float*)d_in[0];
  const int*   pos = (const int*)d_in[1];
  const float* ker = (const float*)d_in[2];
  float*       out = (float*)d_out;

  dim3 grid(BATCH * G * (LSEQ / ROWS_PB));   // 4 * 16 * 32 = 2048 blocks
  masked_gconv_wmma<<<grid, 256, LDS_BYTES, stream>>>(inp, pos, ker, out);
}